// EEACSF_57750130262466
// MI455X (gfx1250) — compile-verified
//
#include <hip/hip_runtime.h>
#include <hip/hip_bf16.h>

// ---------------- problem constants (match reference) ----------------
#define N_SPECIES    16
#define DIM_RAD      16
#define DIM_RAD_ANG  8
#define NMAX_P1      5      // NMAX_ANGLE + 1
#define VDIM         2
#define ROW          432    // 16 (onehot) + 256 (radial) + 160 (angular)
#define RAD_OFF      16
#define ANG_OFF      272

// eta = (dim/end)^2
#define ETA_RAD      10.24f                 // (16/5)^2
#define ETA_ANG      5.2244897959183675f    // (8/3.5)^2
#define STEP_RAD     (5.0f / 15.0f)         // linspace(0,5,16) spacing
#define STEP_ANG     0.5f                   // linspace(0,3.5,8) spacing
#define LOG2E        1.4426950408889634f

typedef __attribute__((ext_vector_type(2))) float v2f;
typedef __attribute__((ext_vector_type(8))) float v8f;

// Hardware f32 global atomic add (global_atomic_add_f32, no return -> STOREcnt).
static __device__ __forceinline__ void atomAddF(float* p, float v) {
    unsafeAtomicAdd(p, v);
}

// ---------------------------------------------------------------------
// Kernel 1: initialize output: onehot block + zeros elsewhere.
// ---------------------------------------------------------------------
__global__ void aev_init_kernel(const int* __restrict__ species,
                                float* __restrict__ out, int total) {
    int i = blockIdx.x * blockDim.x + threadIdx.x;
    if (i >= total) return;
    int a = i / ROW;
    int c = i - a * ROW;
    float v = 0.0f;
    if (c < N_SPECIES) v = (species[a] == c) ? 1.0f : 0.0f;
    out[i] = v;
}

// ---------------------------------------------------------------------
// Kernel 2: radial AEV via V_WMMA_F32_16X16X4_F32.
// One wave = 16 edges x 16 centers per WMMA.
//   exponent[e,r] = (-eta*(d-c_r)^2 + ln(sw)) * log2e   as  A(16x4) x B(4x16)
//   A[e,:] = [d^2, d, 1, log2(sw_e)]
//   B[:,r] = [-eta*log2e, 2*eta*c_r*log2e, -eta*c_r^2*log2e, 1]
//   rt[e,r] = exp2(D[e,r])   (v_exp_f32)
// The quadratic runs on the matrix pipe (co-executes with VALU), VALU does
// exp2 + scatter addressing, memory does 16 atomics/edge (the real bound).
//
// A layout (f32 16x4): lanes 0-15 = {K0,K1} = {d^2, d}; lanes 16-31 = {K2,K3}
//                      = {1, log2 sw}.   B mirrors with K = vgpr (+2 hi half).
// D layout: VGPR v, lane l -> edge (v + 8*(l>=16)), center (l&15).
// ---------------------------------------------------------------------
__global__ void aev_radial_wmma_kernel(const float* __restrict__ dist,
                                       const float* __restrict__ sw,
                                       const int*   __restrict__ esrc,
                                       const int*   __restrict__ edst,
                                       const int*   __restrict__ species,
                                       float* __restrict__ out, int n16) {
    int tid  = blockIdx.x * blockDim.x + threadIdx.x;
    int wave = tid >> 5;            // one 16-edge group per wave32
    int lane = tid & 31;
    int e0   = wave * 16;
    if (e0 >= n16) return;          // wave-uniform: EXEC stays all-ones

    int  sub = lane & 15;
    bool lo  = lane < 16;
    int  idx = e0 + sub;            // edge index carried by this lane (x2)

    float d  = __builtin_nontemporal_load(&dist[idx]);
    float s  = __builtin_nontemporal_load(&sw[idx]);
    int   a  = __builtin_nontemporal_load(&esrc[idx]);
    int   bd = __builtin_nontemporal_load(&edst[idx]);
    int   sd = species[bd];
    // byte offset of out[a, RAD_OFF + sd]; +r*64B per center later
    int   off = (a * ROW + RAD_OFF + sd) * 4;

    v2f A;
    A.x = lo ? d * d : 1.0f;                    // K0 | K2
    A.y = lo ? d     : __log2f(s);              // K1 | K3

    float c  = (float)sub * STEP_RAD;           // center for column N=sub
    float el = ETA_RAD * LOG2E;
    v2f B;
    B.x = lo ? -el            : -el * c * c;    // K0 | K2
    B.y = lo ? 2.0f * el * c  : 1.0f;           // K1 | K3

    v8f C = {};
    // (neg_a, A, neg_b, B, c_mod, C, reuse_a, reuse_b)
    v8f D = __builtin_amdgcn_wmma_f32_16x16x4_f32(false, A, false, B,
                                                  (short)0, C, false, false);

    int rb = sub * (N_SPECIES * 4);             // +r*64 bytes (center stride)
#pragma unroll
    for (int v = 0; v < 8; ++v) {
        float rt   = __builtin_amdgcn_exp2f(D[v]);
        int  offlo = __builtin_amdgcn_readlane(off, v);      // edge v
        int  offhi = __builtin_amdgcn_readlane(off, v + 8);  // edge v+8
        int  o     = (lo ? offlo : offhi) + rb;
        atomAddF((float*)((char*)out + o), rt);
    }
}

// Scalar tail (edges [start, n)) — only used if n_edges % 16 != 0.
__global__ void aev_radial_tail_kernel(const float* __restrict__ dist,
                                       const float* __restrict__ sw,
                                       const int*   __restrict__ esrc,
                                       const int*   __restrict__ edst,
                                       const int*   __restrict__ species,
                                       float* __restrict__ out,
                                       int start, int n_edges) {
    int e = start + blockIdx.x * blockDim.x + threadIdx.x;
    if (e >= n_edges) return;
    float d = dist[e], s = sw[e];
    int a = esrc[e];
    int sd = species[edst[e]];
    float* base = out + (size_t)a * ROW + RAD_OFF + sd;
#pragma unroll
    for (int r = 0; r < DIM_RAD; ++r) {
        float t = d - (float)r * STEP_RAD;
        atomAddF(base + r * N_SPECIES, __expf(-ETA_RAD * t * t) * s);
    }
}

// ---------------------------------------------------------------------
// Kernel 3: angular AEV. One thread per angle (2M threads, 62.5K waves).
// 40-term (gaussian x chebyshev-cos) outer 4-term valence outer, scattered
// with 160 L2-resident global_atomic_add_f32 off one base + immediates.
// ---------------------------------------------------------------------
__global__ void aev_angular_kernel(const float* __restrict__ angles,
                                   const float* __restrict__ dang,
                                   const float* __restrict__ swang,
                                   const int*   __restrict__ central,
                                   const int*   __restrict__ asrc,
                                   const int*   __restrict__ adst,
                                   const int*   __restrict__ edang,
                                   const int*   __restrict__ species,
                                   const float* __restrict__ vtab,
                                   float* __restrict__ out, int n_angles) {
    __shared__ float vt[N_SPECIES * VDIM];  // 32 floats staged once per block
    if (threadIdx.x < N_SPECIES * VDIM) vt[threadIdx.x] = vtab[threadIdx.x];
    __syncthreads();

    int t = blockIdx.x * blockDim.x + threadIdx.x;
    if (t >= n_angles) return;

    float th = __builtin_nontemporal_load(&angles[t]);
    int   ca = __builtin_nontemporal_load(&central[t]);
    int   e1 = __builtin_nontemporal_load(&asrc[t]);
    int   e2 = __builtin_nontemporal_load(&adst[t]);

    // gathers over 500K-edge arrays: cached (hot in L2/L0)
    float d1 = dang[e1],  d2 = dang[e2];
    float s12 = swang[e1] * swang[e2];
    int   a1 = edang[e1], a2 = edang[e2];
    int   sp1 = species[a1], sp2 = species[a2];

    float v10 = vt[sp1 * 2], v11 = vt[sp1 * 2 + 1];
    float v20 = vt[sp2 * 2], v21 = vt[sp2 * 2 + 1];
    float p0 = v10 + v20, p1 = v11 + v21;
    float m0 = v10 * v20, m1 = v11 * v21;
    float val0 = p0 * m0, val1 = p0 * m1, val2 = p1 * m0, val3 = p1 * m1;

    // factor2: 8-term Gaussian basis of d12
    float d12 = 0.5f * (d1 + d2);
    float f2[DIM_RAD_ANG];
#pragma unroll
    for (int ra = 0; ra < DIM_RAD_ANG; ++ra) {
        float dd = d12 - (float)ra * STEP_ANG;
        f2[ra] = __expf(-ETA_ANG * dd * dd);
    }

    // factor1: cos(n*theta)*switch12, n = 0..4 (one v_cos + Chebyshev)
    float c1 = __cosf(th);
    float f1[NMAX_P1];
    f1[0] = s12;
    f1[1] = c1 * s12;
    float cp = c1, cpp = 1.0f;
#pragma unroll
    for (int nn = 2; nn < NMAX_P1; ++nn) {
        float cc = 2.0f * c1 * cp - cpp;
        cpp = cp; cp = cc;
        f1[nn] = cc * s12;
    }

    float* base = out + (size_t)ca * ROW + ANG_OFF;
#pragma unroll
    for (int ra = 0; ra < DIM_RAD_ANG; ++ra) {
#pragma unroll
        for (int f = 0; f < NMAX_P1; ++f) {
            float g = f2[ra] * f1[f];
            float* b = base + (ra * NMAX_P1 + f) * 4;
            atomAddF(b + 0, g * val0);
            atomAddF(b + 1, g * val1);
            atomAddF(b + 2, g * val2);
            atomAddF(b + 3, g * val3);
        }
    }
}

// ---------------------------------------------------------------------
extern "C" void kernel_launch(void* const* d_in, const int* in_sizes, int n_in,
                              void* d_out, int out_size, void* d_ws, size_t ws_size,
                              hipStream_t stream) {
    const int*   species  = (const int*)  d_in[0];
    const float* dist     = (const float*)d_in[1];
    const float* sw       = (const float*)d_in[2];
    const int*   esrc     = (const int*)  d_in[3];
    const int*   edst     = (const int*)  d_in[4];
    const float* angles   = (const float*)d_in[5];
    const float* dang     = (const float*)d_in[6];
    const float* swang    = (const float*)d_in[7];
    const int*   central  = (const int*)  d_in[8];
    const int*   asrc     = (const int*)  d_in[9];
    const int*   adst     = (const int*)  d_in[10];
    const int*   edang    = (const int*)  d_in[11];
    const float* vtab     = (const float*)d_in[12];
    float* out = (float*)d_out;

    const int n_edges  = in_sizes[1];
    const int n_angles = in_sizes[5];
    const int total    = out_size;         // n_atoms * 432

    const int B = 256;                     // 8 wave32 per block
    aev_init_kernel<<<(total + B - 1) / B, B, 0, stream>>>(species, out, total);

    const int n16 = n_edges & ~15;         // 16 edges per wave32
    if (n16 > 0) {
        const int waves  = n16 / 16;
        const int blocks = (waves * 32 + B - 1) / B;
        aev_radial_wmma_kernel<<<blocks, B, 0, stream>>>(
            dist, sw, esrc, edst, species, out, n16);
    }
    if (n_edges > n16) {
        aev_radial_tail_kernel<<<1, B, 0, stream>>>(
            dist, sw, esrc, edst, species, out, n16, n_edges);
    }

    aev_angular_kernel<<<(n_angles + B - 1) / B, B, 0, stream>>>(
        angles, dang, swang, central, asrc, adst, edang, species, vtab,
        out, n_angles);
}